// BeamSearch_50508815400977
// MI455X (gfx1250) — compile-verified
//
#include <hip/hip_runtime.h>
#include <hip/hip_bf16.h>
#include <stdint.h>

#define VOCAB   50257
#define BEAM    8
#define BSZ     64
#define STEPDIM 16
#define NGRAM   4
#define TOPK    16
#define SPLIT   8        // blocks per batch row (== BEAM so a block spans <= 2 beams)
#define TPB     256      // 8 wave32s
#define DEPTH   4        // async LDS ring depth
#define TILE_VEC    256  // B128 loads per tile (one per thread)
#define TILE_FLOATS (TILE_VEC * 4)

__device__ __forceinline__ bool kbetter(float v, int i, float v2, int i2) {
    // total order: value descending, flat index ascending on ties (matches lax.top_k)
    return (v > v2) || ((v == v2) && (i < i2));
}

struct TopK {
    float v[TOPK];
    int   ix[TOPK];
    __device__ __forceinline__ void init() {
#pragma unroll
        for (int k = 0; k < TOPK; ++k) { v[k] = -__builtin_inff(); ix[k] = 0x7fffffff; }
    }
    __device__ __forceinline__ void push(float val, int idx) {
        if (kbetter(val, idx, v[TOPK - 1], ix[TOPK - 1])) {
            v[TOPK - 1] = val; ix[TOPK - 1] = idx;
#pragma unroll
            for (int k = TOPK - 1; k > 0; --k) {
                if (kbetter(v[k], ix[k], v[k - 1], ix[k - 1])) {
                    float tf = v[k];  v[k]  = v[k - 1];  v[k - 1]  = tf;
                    int   ti = ix[k]; ix[k] = ix[k - 1]; ix[k - 1] = ti;
                }
            }
        }
    }
};

__device__ __forceinline__ void issue_async_tile(
    unsigned ringBase, unsigned long long rowBase,
    int t, int vecBeg, int numVec, int tid)
{
    int vi = vecBeg + t * TILE_VEC + tid;
    int vc = (vi < numVec - 1) ? vi : (numVec - 1);
    if (vc < 0) vc = 0;
    unsigned gOff  = ((unsigned)vc) << 4;
    unsigned lAddr = ringBase + ((unsigned)(t % DEPTH)) * (TILE_FLOATS * 4)
                              + (((unsigned)tid) << 4);
    asm volatile("global_load_async_to_lds_b128 %0, %1, %2"
                 :: "v"(lAddr), "v"(gOff), "s"(rowBase) : "memory");
}

__global__ __launch_bounds__(TPB) void topk_stage1(
    const float* __restrict__ lprobs,   // (BSZ, BEAM, VOCAB)
    const float* __restrict__ scores,   // (BSZ, BEAM, STEPDIM)
    const int*   __restrict__ mask,     // (BSZ, BEAM, NGRAM)
    const int*   __restrict__ stepPtr,  // scalar
    float* __restrict__ wsVal,          // (BSZ*SPLIT*TOPK)
    int*   __restrict__ wsIdx)          // (BSZ*SPLIT*TOPK)
{
    __shared__ __align__(16) float ring[DEPTH * TILE_FLOATS];  // 16 KB async staging ring
    __shared__ float sVal[TPB * TOPK];                         // 16 KB
    __shared__ int   sIdx[TPB * TOPK];                         // 16 KB
    __shared__ float l1Val[32 * TOPK];
    __shared__ int   l1Idx[32 * TOPK];
    __shared__ float sBias[BEAM];
    __shared__ int   sKeep[BEAM];

    const int tid  = threadIdx.x;
    const int b    = blockIdx.x / SPLIT;
    const int p    = blockIdx.x % SPLIT;
    const int step = *stepPtr;

    // per-beam keep flag and score bias
    if (tid < BEAM) {
        const int* m = mask + (b * BEAM + tid) * NGRAM;
        int s = m[0] + m[1] + m[2] + m[3];
        sKeep[tid] = (s == NGRAM) ? 1 : 0;
        sBias[tid] = (step > 0) ? scores[(b * BEAM + tid) * STEPDIM + (step - 1)] : 0.0f;
    }
    __syncthreads();

    const int nBeam  = (step == 0) ? 1 : BEAM;
    const int N      = nBeam * VOCAB;          // flat elements per row considered
    const int numVec = (N + 3) >> 2;           // B128 vectors
    const int vecPer = (numVec + SPLIT - 1) / SPLIT;
    const int vecBeg = p * vecPer;
    const int vecEndU = vecBeg + vecPer;
    const int vecEnd = (vecEndU < numVec) ? vecEndU : numVec;
    const int cnt    = (vecEnd > vecBeg) ? (vecEnd - vecBeg) : 0;
    const int numTiles = (cnt + TILE_VEC - 1) / TILE_VEC;

    // This block's element range spans at most two beams (SPLIT == BEAM):
    // hoist both beams' (keep-mask, bias) into registers; inner loop then
    // needs only a compare against the beam boundary — no division, no LDS.
    int beamLo = (vecBeg * 4) / VOCAB;
    if (beamLo > nBeam - 1) beamLo = nBeam - 1;
    int beamHi = (beamLo + 1 < nBeam) ? beamLo + 1 : beamLo;
    const int      boundary = (beamLo + 1) * VOCAB;
    const unsigned mLo = sKeep[beamLo] ? 0xffffffffu : 0u;
    const unsigned mHi = sKeep[beamHi] ? 0xffffffffu : 0u;
    const float    bLo = sBias[beamLo];
    const float    bHi = sBias[beamHi];

    const unsigned long long rowBase =
        (unsigned long long)(uintptr_t)(lprobs + (size_t)b * (size_t)(BEAM * VOCAB));
    const unsigned ringBase = (unsigned)(uintptr_t)(&ring[0]);

    TopK tk; tk.init();

    // --- async global -> LDS pipeline -------------------------------------
    // Each thread loads/consumes exactly its own 16B of each 4KB tile, so a
    // re-issue into the just-consumed ring slot only overwrites data this
    // thread has already consumed: one barrier per tile suffices.
    int issued = (numTiles < DEPTH) ? numTiles : DEPTH;
    for (int t0 = 0; t0 < issued; ++t0)
        issue_async_tile(ringBase, rowBase, t0, vecBeg, numVec, tid);

    int t = 0;

    // steady state: always DEPTH tiles in flight -> constant wait immediate
    for (; issued < numTiles; ++t) {
        asm volatile("s_wait_asynccnt 3" ::: "memory");  // DEPTH-1
        __syncthreads();

        int vi = vecBeg + t * TILE_VEC + tid;
        if (vi < vecEnd) {
            const float4 x =
                *reinterpret_cast<const float4*>(&ring[(t % DEPTH) * TILE_FLOATS + (tid << 2)]);
            float xa[4] = {x.x, x.y, x.z, x.w};
            int base = vi << 2;
#pragma unroll
            for (int j = 0; j < 4; ++j) {
                int fi = base + j;
                bool hiSel = fi >= boundary;
                unsigned msk = hiSel ? mHi : mLo;
                float bias   = hiSel ? bHi : bLo;
                float vv = __uint_as_float(__float_as_uint(xa[j]) & msk) + bias;
                if (fi < N) tk.push(vv, fi);
            }
        }
        issue_async_tile(ringBase, rowBase, issued, vecBeg, numVec, tid);
        ++issued;
    }

    // drain: 0..DEPTH-1 tiles left in flight
    for (; t < numTiles; ++t) {
        switch (numTiles - t - 1) {
            case 0:  asm volatile("s_wait_asynccnt 0" ::: "memory"); break;
            case 1:  asm volatile("s_wait_asynccnt 1" ::: "memory"); break;
            case 2:  asm volatile("s_wait_asynccnt 2" ::: "memory"); break;
            default: asm volatile("s_wait_asynccnt 3" ::: "memory"); break;
        }
        __syncthreads();

        int vi = vecBeg + t * TILE_VEC + tid;
        if (vi < vecEnd) {
            const float4 x =
                *reinterpret_cast<const float4*>(&ring[(t % DEPTH) * TILE_FLOATS + (tid << 2)]);
            float xa[4] = {x.x, x.y, x.z, x.w};
            int base = vi << 2;
#pragma unroll
            for (int j = 0; j < 4; ++j) {
                int fi = base + j;
                bool hiSel = fi >= boundary;
                unsigned msk = hiSel ? mHi : mLo;
                float bias   = hiSel ? bHi : bLo;
                float vv = __uint_as_float(__float_as_uint(xa[j]) & msk) + bias;
                if (fi < N) tk.push(vv, fi);
            }
        }
    }

    // --- block-level reduction of 256 sorted lists ------------------------
#pragma unroll
    for (int k = 0; k < TOPK; ++k) { sVal[tid * TOPK + k] = tk.v[k]; sIdx[tid * TOPK + k] = tk.ix[k]; }
    __syncthreads();

    if (tid < 32) {
        TopK m; m.init();
        int beg = tid * (TPB / 32) * TOPK;   // 8 lists = 128 candidates
        for (int j = 0; j < (TPB / 32) * TOPK; ++j) m.push(sVal[beg + j], sIdx[beg + j]);
#pragma unroll
        for (int k = 0; k < TOPK; ++k) { l1Val[tid * TOPK + k] = m.v[k]; l1Idx[tid * TOPK + k] = m.ix[k]; }
    }
    __syncthreads();

    if (tid == 0) {
        TopK f; f.init();
        for (int j = 0; j < 32 * TOPK; ++j) f.push(l1Val[j], l1Idx[j]);
        int o = (b * SPLIT + p) * TOPK;
#pragma unroll
        for (int k = 0; k < TOPK; ++k) { wsVal[o + k] = f.v[k]; wsIdx[o + k] = f.ix[k]; }
    }
}

__global__ __launch_bounds__(32) void topk_stage2(
    const float* __restrict__ wsVal,
    const int*   __restrict__ wsIdx,
    float* __restrict__ out)            // [scores 1024 | indices 1024 | beams 1024]
{
    int row = blockIdx.x * blockDim.x + threadIdx.x;
    if (row >= BSZ) return;
    TopK f; f.init();
    int beg = row * SPLIT * TOPK;
    for (int j = 0; j < SPLIT * TOPK; ++j) f.push(wsVal[beg + j], wsIdx[beg + j]);
#pragma unroll
    for (int k = 0; k < TOPK; ++k) {
        unsigned idx = (unsigned)f.ix[k];
        out[row * TOPK + k]                  = f.v[k];
        out[BSZ * TOPK + row * TOPK + k]     = (float)(idx % (unsigned)VOCAB);
        out[2 * BSZ * TOPK + row * TOPK + k] = (float)(idx / (unsigned)VOCAB);
    }
}

extern "C" void kernel_launch(void* const* d_in, const int* in_sizes, int n_in,
                              void* d_out, int out_size, void* d_ws, size_t ws_size,
                              hipStream_t stream) {
    const float* lprobs  = (const float*)d_in[0];
    const float* scores  = (const float*)d_in[1];
    const int*   mask    = (const int*)d_in[2];
    const int*   stepPtr = (const int*)d_in[3];

    float* wsVal = (float*)d_ws;                                         // 64*8*16 floats
    int*   wsIdx = (int*)((char*)d_ws + (size_t)BSZ * SPLIT * TOPK * sizeof(float));

    topk_stage1<<<BSZ * SPLIT, TPB, 0, stream>>>(lprobs, scores, mask, stepPtr, wsVal, wsIdx);
    topk_stage2<<<(BSZ + 31) / 32, 32, 0, stream>>>(wsVal, wsIdx, (float*)d_out);
}